// Fusion_1778116461036
// MI455X (gfx1250) — compile-verified
//
#include <hip/hip_runtime.h>
#include <math.h>

// out = beta1*v1 + beta2*v2 with per-row betas from top-2 statistics of prob rows.
// Shapes: v1,v2: (N,128) f32; prob_v1,prob_v2: (N,10) f32; out: (N,128) f32.
// Pure streaming kernel: ~808 MB total traffic, roofline ~35us @ 23.3 TB/s.

typedef __attribute__((ext_vector_type(4))) float f32x4;

#define LAM   0.9f
#define EPS   1e-8f
#define DVEC  128
#define CPROB 10
#define ROWS_PER_BLOCK 8   // 256 threads = 8 wave32 = 8 rows
#define NEG_BIG (-3.0e38f)

__global__ __launch_bounds__(256)
void weighted_blend_kernel(const float* __restrict__ v1,
                           const float* __restrict__ prob_v1,
                           const float* __restrict__ v2,
                           const float* __restrict__ prob_v2,
                           float* __restrict__ out,
                           int N)
{
    const int lane = threadIdx.x & 31;          // wave32 lane
    const int wid  = threadIdx.x >> 5;          // wave within block
    const int row  = blockIdx.x * ROWS_PER_BLOCK + wid;
    if (row >= N) return;

    // ---- per-row weights, computed once per wave -------------------------
    // Lanes 0..15 handle prob_v1[row], lanes 16..31 handle prob_v2[row].
    const int  l  = lane & 15;
    const bool hi = (lane >= 16);

    float fir = NEG_BIG;
    float sec = NEG_BIG;
    if (l < CPROB) {
        const float* __restrict__ p = hi ? prob_v2 : prob_v1;
        fir = p[(size_t)row * CPROB + l];       // (value, -inf) pair
    }

    // Branch-free top-2 tree reduction within each 16-lane half.
    // xor masks 1,2,4,8 (<16) never cross the half boundary.
    #pragma unroll
    for (int m = 1; m <= 8; m <<= 1) {
        float ofir = __shfl_xor(fir, m, 32);
        float osec = __shfl_xor(sec, m, 32);
        float nfir = fmaxf(fir, ofir);
        float nsec = fmaxf(fminf(fir, ofir), fmaxf(sec, osec));
        fir = nfir;
        sec = nsec;
    }

    // w = exp( LAM*log(fir+eps) + (1-LAM)*log(fir-sec+eps) ), ALPHA == 1
    float w = expf(LAM * logf(fir + EPS) + (1.0f - LAM) * logf(fir - sec + EPS));

    // Swap halves: every lane gets both w1 and w2.
    float wo = __shfl_xor(w, 16, 32);
    float w1 = hi ? wo : w;
    float w2 = hi ? w  : wo;
    float inv = 1.0f / (w1 + w2);
    float b1  = w1 * inv;
    float b2  = w2 * inv;

    // ---- streaming blend: 32 lanes x float4 == 128 floats per row --------
    // Non-temporal: zero reuse on a 768MB stream; keep it out of L2.
    const size_t base = (size_t)row * DVEC + (size_t)lane * 4;
    const f32x4* __restrict__ pa = reinterpret_cast<const f32x4*>(v1 + base);
    const f32x4* __restrict__ pb = reinterpret_cast<const f32x4*>(v2 + base);

    f32x4 a = __builtin_nontemporal_load(pa);   // global_load_b128 TH_NT
    f32x4 b = __builtin_nontemporal_load(pb);

    f32x4 o;
    o.x = fmaf(b1, a.x, b2 * b.x);
    o.y = fmaf(b1, a.y, b2 * b.y);
    o.z = fmaf(b1, a.z, b2 * b.z);
    o.w = fmaf(b1, a.w, b2 * b.w);

    __builtin_nontemporal_store(o, reinterpret_cast<f32x4*>(out + base));
}

extern "C" void kernel_launch(void* const* d_in, const int* in_sizes, int n_in,
                              void* d_out, int out_size, void* d_ws, size_t ws_size,
                              hipStream_t stream)
{
    // setup_inputs() order: v1 (N*128), prob_v1 (N*10), v2 (N*128), prob_v2 (N*10)
    const float* v1 = (const float*)d_in[0];
    const float* p1 = (const float*)d_in[1];
    const float* v2 = (const float*)d_in[2];
    const float* p2 = (const float*)d_in[3];
    float* out = (float*)d_out;

    const int N = in_sizes[0] / DVEC;           // 500000
    const int blocks = (N + ROWS_PER_BLOCK - 1) / ROWS_PER_BLOCK;

    weighted_blend_kernel<<<blocks, 256, 0, stream>>>(v1, p1, v2, p2, out, N);
}